// ExpertParallelWrapper_50371376448121
// MI455X (gfx1250) — compile-verified
//
#include <hip/hip_runtime.h>
#include <hip/hip_bf16.h>
#include <stdint.h>

// ---------------------------------------------------------------------------
// MoE top-2/16 MLP for MI455X (gfx1250, wave32).
// HBM-bound on fp32 weight traffic (~0.6 GB total -> ~25us at 23.3 TB/s),
// so we keep fp32 end-to-end and use V_WMMA_F32_16X16X4_F32 for both GEMMs.
// ---------------------------------------------------------------------------

#define MOE_B     8192
#define MOE_DIN   1024
#define MOE_DHID  2048
#define MOE_DOUT  1024
#define MOE_E     16

#define TILE_M    32          // routed tokens per block
#define KC        128         // K-chunk staged in LDS
#define XPAD      4           // row pad (floats): stride 132 -> 16B aligned, conflict-free
#define MAX_TILES (MOE_B / TILE_M)   // 256 worst-case M-tiles per expert

typedef float v2f __attribute__((ext_vector_type(2)));
typedef float v8f __attribute__((ext_vector_type(8)));

// ---------------------------------------------------------------------------
// Kernel 1: gating. One wave32 per token. logits = x_row @ Wg (1024x16),
// lane-parallel over d, shfl-xor reduce. Renormalized top-2 softmax weight
// collapses to w0 = 1/(1+exp(l2-l1)). Atomic push into per-expert lists.
// ---------------------------------------------------------------------------
__global__ __launch_bounds__(256) void moe_gate(
    const float* __restrict__ x, const float* __restrict__ Wg,
    int* __restrict__ counts, int* __restrict__ tlist, float* __restrict__ wlist)
{
    const int wave = threadIdx.x >> 5;
    const int lane = threadIdx.x & 31;
    const int b = blockIdx.x * 8 + wave;

    float acc[MOE_E];
#pragma unroll
    for (int j = 0; j < MOE_E; ++j) acc[j] = 0.f;

    const float* xrow = x + (size_t)b * MOE_DIN;
    for (int d = lane; d < MOE_DIN; d += 32) {
        const float xv = xrow[d];
        const float* wr = Wg + (size_t)d * MOE_E;
#pragma unroll
        for (int j = 0; j < MOE_E; ++j) acc[j] = fmaf(xv, wr[j], acc[j]);
    }
    // wave32 all-reduce of 16 partial logits
#pragma unroll
    for (int j = 0; j < MOE_E; ++j) {
        float v = acc[j];
#pragma unroll
        for (int off = 16; off >= 1; off >>= 1) v += __shfl_xor(v, off, 32);
        acc[j] = v;
    }
    // top-2 (first occurrence on ties, matching lax.top_k)
    int i1 = 0; float l1 = acc[0];
#pragma unroll
    for (int j = 1; j < MOE_E; ++j) if (acc[j] > l1) { l1 = acc[j]; i1 = j; }
    int i2 = (i1 == 0) ? 1 : 0; float l2 = acc[i2];
#pragma unroll
    for (int j = 0; j < MOE_E; ++j)
        if (j != i1 && acc[j] > l2) { l2 = acc[j]; i2 = j; }

    const float w0 = 1.f / (1.f + __expf(l2 - l1));
    const float w1 = 1.f - w0;

    if (lane == 0) {
        int p0 = atomicAdd(&counts[i1], 1);
        tlist[i1 * MOE_B + p0] = b;  wlist[i1 * MOE_B + p0] = w0;
        int p1 = atomicAdd(&counts[i2], 1);
        tlist[i2 * MOE_B + p1] = b;  wlist[i2 * MOE_B + p1] = w1;
    }
}

// ---------------------------------------------------------------------------
// Kernel 2: 16-element exclusive scan -> per-expert row offsets into H.
// ---------------------------------------------------------------------------
__global__ void moe_scan(const int* __restrict__ counts, int* __restrict__ offsets)
{
    if (threadIdx.x == 0) {
        int s = 0;
        for (int e = 0; e < MOE_E; ++e) { offsets[e] = s; s += counts[e]; }
    }
}

// ---------------------------------------------------------------------------
// GEMM1: H[rows, 2048] = relu( gather(x, tlist) @ W1[e] + b1[e] ).
// Block = 256 threads = 8 waves. Tile: M=32 tokens, N=512 (wave owns 64 cols).
// A staged in LDS per KC chunk; B streamed coalesced from global.
// A-layout per ISA: V0/V1 hold K pairs; lanes 0-15 -> K{0,1}, 16-31 -> K{2,3}.
// ---------------------------------------------------------------------------
__global__ __launch_bounds__(256) void moe_gemm1(
    const float* __restrict__ x, const float* __restrict__ W1,
    const float* __restrict__ b1, const int* __restrict__ counts,
    const int* __restrict__ offsets, const int* __restrict__ tlist,
    float* __restrict__ H)
{
    const int e     = blockIdx.x >> 8;        // blockIdx.x = e*256 + tileM
    const int tileM = blockIdx.x & 255;
    const int cnt   = counts[e];
    if (tileM * TILE_M >= cnt) return;
    const int off_e = offsets[e];

    __shared__ __align__(16) float Xs[TILE_M][KC + XPAD];
    __shared__ int sTok[TILE_M];

    if (threadIdx.x < TILE_M) {
        const int g = tileM * TILE_M + threadIdx.x;
        sTok[threadIdx.x] = (g < cnt) ? tlist[e * MOE_B + g] : 0;
    }
    __syncthreads();

    const int wave  = threadIdx.x >> 5;
    const int lane  = threadIdx.x & 31;
    const int hi    = lane >> 4;              // 0: lanes 0-15, 1: lanes 16-31
    const int ln    = lane & 15;
    const int nbase = blockIdx.y * 512 + wave * 64;

    v8f acc[2][4] = {};

    const int ldr = threadIdx.x >> 3;         // 0..31: LDS load row
    const int ldc = (threadIdx.x & 7) * 4;    // 0..28: LDS load col base

    for (int kc = 0; kc < MOE_DIN; kc += KC) {
        // stage gathered X chunk: 32 rows x 128 floats
        const float* xr = x + (size_t)sTok[ldr] * MOE_DIN + kc;
#pragma unroll
        for (int i = 0; i < 4; ++i) {
            const int c = ldc + i * 32;
            *(float4*)&Xs[ldr][c] = *(const float4*)(xr + c);
        }
        __syncthreads();

        for (int k4 = 0; k4 < KC; k4 += 4) {
            const int koff = k4 + hi * 2;     // lane-group K offset within chunk
            v2f a0, a1;
            a0.x = Xs[ln][koff];       a0.y = Xs[ln][koff + 1];
            a1.x = Xs[16 + ln][koff];  a1.y = Xs[16 + ln][koff + 1];

            const float* wp = W1
                + ((size_t)e * MOE_DIN + (size_t)(kc + k4 + hi * 2)) * MOE_DHID
                + nbase + ln;
#pragma unroll
            for (int nt = 0; nt < 4; ++nt) {
                v2f bv;
                bv.x = wp[nt * 16];
                bv.y = wp[nt * 16 + MOE_DHID];
                acc[0][nt] = __builtin_amdgcn_wmma_f32_16x16x4_f32(
                    false, a0, false, bv, (short)0, acc[0][nt], false, false);
                acc[1][nt] = __builtin_amdgcn_wmma_f32_16x16x4_f32(
                    false, a1, false, bv, (short)0, acc[1][nt], false, false);
            }
        }
        __syncthreads();
    }

    // epilogue: bias + relu, guarded store into compacted H
#pragma unroll
    for (int mt = 0; mt < 2; ++mt) {
#pragma unroll
        for (int nt = 0; nt < 4; ++nt) {
            const int n = nbase + nt * 16 + ln;
            const float bias = b1[e * MOE_DHID + n];
#pragma unroll
            for (int r = 0; r < 8; ++r) {
                const int mrow = mt * 16 + hi * 8 + r;   // C layout: V[r] -> M=r / M=8+r
                const int g = tileM * TILE_M + mrow;
                if (g < cnt) {
                    float h = acc[mt][nt][r] + bias;
                    H[(size_t)(off_e + g) * MOE_DHID + n] = fmaxf(h, 0.f);
                }
            }
        }
    }
}

// ---------------------------------------------------------------------------
// GEMM2: out[tok] += w * ( H_rows @ W2[e] + b2[e] ).  Same tiling, K=2048.
// H rows are contiguous per expert (compacted), so staging loads coalesce.
// Scatter uses global_atomic_add_f32 (two experts may hit the same token).
// ---------------------------------------------------------------------------
__global__ __launch_bounds__(256) void moe_gemm2(
    const float* __restrict__ H, const float* __restrict__ W2,
    const float* __restrict__ b2, const int* __restrict__ counts,
    const int* __restrict__ offsets, const int* __restrict__ tlist,
    const float* __restrict__ wlist, float* __restrict__ out)
{
    const int e     = blockIdx.x >> 8;
    const int tileM = blockIdx.x & 255;
    const int cnt   = counts[e];
    if (tileM * TILE_M >= cnt) return;
    const int off_e = offsets[e];

    __shared__ __align__(16) float Hs[TILE_M][KC + XPAD];
    __shared__ int   sTok[TILE_M];
    __shared__ float sW[TILE_M];
    __shared__ int   sRow[TILE_M];

    if (threadIdx.x < TILE_M) {
        const int g = tileM * TILE_M + threadIdx.x;
        const bool v = (g < cnt);
        sTok[threadIdx.x] = v ? tlist[e * MOE_B + g] : 0;
        sW[threadIdx.x]   = v ? wlist[e * MOE_B + g] : 0.f;
        sRow[threadIdx.x] = off_e + (v ? g : 0);   // clamp pad rows in-range
    }
    __syncthreads();

    const int wave  = threadIdx.x >> 5;
    const int lane  = threadIdx.x & 31;
    const int hi    = lane >> 4;
    const int ln    = lane & 15;
    const int nbase = blockIdx.y * 512 + wave * 64;

    v8f acc[2][4] = {};

    const int ldr = threadIdx.x >> 3;
    const int ldc = (threadIdx.x & 7) * 4;

    for (int kc = 0; kc < MOE_DHID; kc += KC) {
        const float* hr = H + (size_t)sRow[ldr] * MOE_DHID + kc;
#pragma unroll
        for (int i = 0; i < 4; ++i) {
            const int c = ldc + i * 32;
            *(float4*)&Hs[ldr][c] = *(const float4*)(hr + c);
        }
        __syncthreads();

        for (int k4 = 0; k4 < KC; k4 += 4) {
            const int koff = k4 + hi * 2;
            v2f a0, a1;
            a0.x = Hs[ln][koff];       a0.y = Hs[ln][koff + 1];
            a1.x = Hs[16 + ln][koff];  a1.y = Hs[16 + ln][koff + 1];

            const float* wp = W2
                + ((size_t)e * MOE_DHID + (size_t)(kc + k4 + hi * 2)) * MOE_DOUT
                + nbase + ln;
#pragma unroll
            for (int nt = 0; nt < 4; ++nt) {
                v2f bv;
                bv.x = wp[nt * 16];
                bv.y = wp[nt * 16 + MOE_DOUT];
                acc[0][nt] = __builtin_amdgcn_wmma_f32_16x16x4_f32(
                    false, a0, false, bv, (short)0, acc[0][nt], false, false);
                acc[1][nt] = __builtin_amdgcn_wmma_f32_16x16x4_f32(
                    false, a1, false, bv, (short)0, acc[1][nt], false, false);
            }
        }
        __syncthreads();
    }

    // epilogue: bias, routing weight, atomic scatter-accumulate
#pragma unroll
    for (int mt = 0; mt < 2; ++mt) {
#pragma unroll
        for (int nt = 0; nt < 4; ++nt) {
            const int n = nbase + nt * 16 + ln;
            const float bias = b2[e * MOE_DOUT + n];
#pragma unroll
            for (int r = 0; r < 8; ++r) {
                const int mrow = mt * 16 + hi * 8 + r;
                const int g = tileM * TILE_M + mrow;
                if (g < cnt) {
                    const float yv = acc[mt][nt][r] + bias;
                    atomicAdd(&out[(size_t)sTok[mrow] * MOE_DOUT + n], sW[mrow] * yv);
                }
            }
        }
    }
}

// ---------------------------------------------------------------------------
// Launch. Inputs: x, Wg, W1, b1, W2, b2 (all fp32). Output: [B, D_OUT] fp32.
// Workspace layout (needs ~129.1 MB): counts | offsets | tlist | wlist | H
// ---------------------------------------------------------------------------
extern "C" void kernel_launch(void* const* d_in, const int* in_sizes, int n_in,
                              void* d_out, int out_size, void* d_ws, size_t ws_size,
                              hipStream_t stream)
{
    const float* x  = (const float*)d_in[0];
    const float* Wg = (const float*)d_in[1];
    const float* W1 = (const float*)d_in[2];
    const float* b1 = (const float*)d_in[3];
    const float* W2 = (const float*)d_in[4];
    const float* b2 = (const float*)d_in[5];
    float* out = (float*)d_out;

    uintptr_t p = (uintptr_t)d_ws;
    int*   counts  = (int*)p;    p += 256;
    int*   offsets = (int*)p;    p += 256;
    int*   tlist   = (int*)p;    p += (size_t)MOE_E * MOE_B * sizeof(int);
    float* wlist   = (float*)p;  p += (size_t)MOE_E * MOE_B * sizeof(float);
    float* H       = (float*)p;  // (B*K) x D_HID fp32 = 128 MB

    hipMemsetAsync(out, 0, (size_t)MOE_B * MOE_DOUT * sizeof(float), stream);
    hipMemsetAsync(counts, 0, MOE_E * sizeof(int), stream);

    moe_gate<<<MOE_B / 8, 256, 0, stream>>>(x, Wg, counts, tlist, wlist);
    moe_scan<<<1, 32, 0, stream>>>(counts, offsets);
    moe_gemm1<<<dim3(MOE_E * MAX_TILES, MOE_DHID / 512), 256, 0, stream>>>(
        x, W1, b1, counts, offsets, tlist, H);
    moe_gemm2<<<dim3(MOE_E * MAX_TILES, MOE_DOUT / 512), 256, 0, stream>>>(
        H, W2, b2, counts, offsets, tlist, wlist, out);
}